// EntropySC_79517024518423
// MI455X (gfx1250) — compile-verified
//
#include <hip/hip_runtime.h>
#include <math.h>

// ---------------------------------------------------------------------------
// EntropySC on MI455X (gfx1250).
// Phase 1: zero lookup table (workspace)       -- 33.5 MB, L2-resident
// Phase 2: scatter entropy values per CA atom  -- racy like JAX .at[].set
// Phase 3: streaming elementwise select        -- async-to-LDS pipelined
// ---------------------------------------------------------------------------

typedef float f32x4 __attribute__((ext_vector_type(4)));

#define TEMPERATURE 298.0f
#define PAD_INDEX   20
#define CA_ID       1
#define DIM_B 64
#define DIM_C 4
#define DIM_R 4096
#define DIM_A 8

#define EL_THREADS 256   // 8 wave32s per block
#define EL_ITERS   8     // float4 chunks per thread

// ---- CDNA5 async copy: 16B global -> LDS, tracked by ASYNCcnt --------------
__device__ __forceinline__ void async_ld16(unsigned lds_addr,
                                           unsigned long long gaddr) {
  asm volatile("global_load_async_to_lds_b128 %0, %1, off"
               :: "v"(lds_addr), "v"(gaddr)
               : "memory");
}

__device__ __forceinline__ unsigned lds_of(const void* p) {
  // generic pointer to LDS: low 32 bits are the wave-relative LDS offset
  return (unsigned)(unsigned long long)p;
}

// ---------------------------------------------------------------------------
// Kernel 1: zero the lookup_sc scratch table
// ---------------------------------------------------------------------------
__global__ void zero_ws_kernel(f32x4* __restrict__ ws, int n4) {
  int i = blockIdx.x * blockDim.x + threadIdx.x;
  if (i < n4) {
    f32x4 z = {0.f, 0.f, 0.f, 0.f};
    ws[i] = z;
  }
}

// ---------------------------------------------------------------------------
// Kernel 2: scatter entropy values (one thread per atom)
// ---------------------------------------------------------------------------
__global__ void scatter_kernel(const int* __restrict__ ad,            // (N,5)
                               const unsigned char* __restrict__ alt, // (N,8) bool
                               const float* __restrict__ weight,      // (1,)
                               const float* __restrict__ table,       // (21,)
                               float* __restrict__ lookup,            // (B,C,R,A)
                               int n_atoms) {
  int i = blockIdx.x * blockDim.x + threadIdx.x;
  if (i >= n_atoms) return;

  long long row = (long long)i * 5;
  int at = ad[row + 0];
  int rn = ad[row + 1];
  if (at != CA_ID || rn == PAD_INDEX) return;

  int b = ad[row + 2];
  int c = ad[row + 3];
  int r = ad[row + 4];

  int rc = rn < 0 ? 0 : (rn > PAD_INDEX ? PAD_INDEX : rn);
  float scale = (1.0f - tanhf(-weight[0])) * TEMPERATURE;
  float val   = table[rc] * scale;

  long long base = ((((long long)b * DIM_C + c) * DIM_R) + r) * DIM_A;
  long long abase = (long long)i * DIM_A;
#pragma unroll
  for (int a = 0; a < DIM_A; ++a) {
    if (alt[abase + a]) lookup[base + a] = val;  // duplicate winner unspecified (matches JAX)
  }
}

// ---------------------------------------------------------------------------
// Kernel 3: streaming elementwise select, async-to-LDS double-buffered.
// Each thread stages its own 5 x float4 into private LDS slots; wave-level
// s_wait_asynccnt is sufficient (no cross-thread sharing -> no barrier).
// ---------------------------------------------------------------------------
__global__ void fuse_kernel(const f32x4* __restrict__ saSC,
                            const f32x4* __restrict__ hbond,
                            const f32x4* __restrict__ vdw,
                            const f32x4* __restrict__ electro,
                            const f32x4* __restrict__ lookup,
                            f32x4* __restrict__ out) {
  __shared__ f32x4 sb[2][5][EL_THREADS];   // 40 KB, double buffered

  const int tid = threadIdx.x;
  const long long base = (long long)blockIdx.x * (EL_THREADS * EL_ITERS) + tid;

  // stage iteration 0 into buffer 0
  {
    long long idx = base;
    async_ld16(lds_of(&sb[0][0][tid]), (unsigned long long)(const void*)(saSC    + idx));
    async_ld16(lds_of(&sb[0][1][tid]), (unsigned long long)(const void*)(hbond   + idx));
    async_ld16(lds_of(&sb[0][2][tid]), (unsigned long long)(const void*)(vdw     + idx));
    async_ld16(lds_of(&sb[0][3][tid]), (unsigned long long)(const void*)(electro + idx));
    async_ld16(lds_of(&sb[0][4][tid]), (unsigned long long)(const void*)(lookup  + idx));
  }

  int cur = 0;
  for (int it = 0; it < EL_ITERS; ++it) {
    if (it + 1 < EL_ITERS) {
      // prefetch next tile into the other buffer, then wait for current tile
      long long idx = base + (long long)(it + 1) * EL_THREADS;
      int nxt = cur ^ 1;
      async_ld16(lds_of(&sb[nxt][0][tid]), (unsigned long long)(const void*)(saSC    + idx));
      async_ld16(lds_of(&sb[nxt][1][tid]), (unsigned long long)(const void*)(hbond   + idx));
      async_ld16(lds_of(&sb[nxt][2][tid]), (unsigned long long)(const void*)(vdw     + idx));
      async_ld16(lds_of(&sb[nxt][3][tid]), (unsigned long long)(const void*)(electro + idx));
      async_ld16(lds_of(&sb[nxt][4][tid]), (unsigned long long)(const void*)(lookup  + idx));
      asm volatile("s_wait_asynccnt 0x5" ::: "memory"); // <=5 outstanding = current tile done
    } else {
      asm volatile("s_wait_asynccnt 0x0" ::: "memory");
    }

    f32x4 s = sb[cur][0][tid];
    f32x4 h = sb[cur][1][tid];
    f32x4 v = sb[cur][2][tid];
    f32x4 e = sb[cur][3][tid];
    f32x4 l = sb[cur][4][tid];

    f32x4 o;
#pragma unroll
    for (int k = 0; k < 4; ++k) {
      float lv = l[k], sv = s[k], hv = h[k], vv = v[k], ev = e[k];
      float fin  = lv * fmaxf(sv, 0.f);
      float corr = (ev > 0.f) ? 0.2f : 1.0f;
      float re   = fabsf(hv + vv + ev * corr);
      o[k] = (lv < re) ? lv : ((fin < re) ? re : fin);
    }

    // write-once output: non-temporal so it doesn't rinse L2 (keep lookup_sc hot)
    __builtin_nontemporal_store(o, &out[base + (long long)it * EL_THREADS]);
    cur ^= 1;
  }
}

// ---------------------------------------------------------------------------
// Scalar tail (only runs if sizes are not multiples of the tile; with the
// reference shapes it is never launched)
// ---------------------------------------------------------------------------
__global__ void fuse_tail_kernel(const float* __restrict__ saSC,
                                 const float* __restrict__ hbond,
                                 const float* __restrict__ vdw,
                                 const float* __restrict__ electro,
                                 const float* __restrict__ lookup,
                                 float* __restrict__ out,
                                 long long start, long long n) {
  long long i = start + (long long)blockIdx.x * blockDim.x + threadIdx.x;
  if (i >= n) return;
  float lv = lookup[i], sv = saSC[i], hv = hbond[i], vv = vdw[i], ev = electro[i];
  float fin  = lv * fmaxf(sv, 0.f);
  float corr = (ev > 0.f) ? 0.2f : 1.0f;
  float re   = fabsf(hv + vv + ev * corr);
  out[i] = (lv < re) ? lv : ((fin < re) ? re : fin);
}

extern "C" void kernel_launch(void* const* d_in, const int* in_sizes, int n_in,
                              void* d_out, int out_size, void* d_ws, size_t ws_size,
                              hipStream_t stream) {
  // setup_inputs() order:
  const int*           ad      = (const int*)d_in[0];            // (N,5) int32
  const float*         saSC    = (const float*)d_in[1];          // (B,C,R,A) f32
  const float*         hbond   = (const float*)d_in[2];
  const float*         vdw     = (const float*)d_in[3];
  const float*         electro = (const float*)d_in[4];
  const unsigned char* alt     = (const unsigned char*)d_in[5];  // (N,8) bool
  const float*         weight  = (const float*)d_in[6];          // (1,)
  const float*         table   = (const float*)d_in[7];          // (21,)
  float*               out     = (float*)d_out;
  float*               lookup  = (float*)d_ws;                   // 33.5 MB scratch

  const int n_atoms = in_sizes[0] / 5;
  const int ntot    = out_size;          // B*C*R*A = 8,388,608
  const int n4      = ntot / 4;

  // Phase 1: zero lookup table (must re-zero every call for determinism)
  zero_ws_kernel<<<(n4 + EL_THREADS - 1) / EL_THREADS, EL_THREADS, 0, stream>>>(
      (f32x4*)lookup, n4);

  // Phase 2: scatter
  scatter_kernel<<<(n_atoms + EL_THREADS - 1) / EL_THREADS, EL_THREADS, 0, stream>>>(
      ad, alt, weight, table, lookup, n_atoms);

  // Phase 3: streaming map (async-to-LDS pipelined), 2048 f32x4 per block
  const int per_block = EL_THREADS * EL_ITERS;
  const int blocks    = n4 / per_block;
  if (blocks > 0) {
    fuse_kernel<<<blocks, EL_THREADS, 0, stream>>>(
        (const f32x4*)saSC, (const f32x4*)hbond, (const f32x4*)vdw,
        (const f32x4*)electro, (const f32x4*)lookup, (f32x4*)out);
  }
  const long long done = (long long)blocks * per_block * 4;
  const long long rem  = (long long)ntot - done;
  if (rem > 0) {
    fuse_tail_kernel<<<(int)((rem + EL_THREADS - 1) / EL_THREADS), EL_THREADS, 0, stream>>>(
        saSC, hbond, vdw, electro, lookup, out, done, (long long)ntot);
  }
}